// CRFRNN_48326972014746
// MI455X (gfx1250) — compile-verified
//
#include <hip/hip_runtime.h>
#include <math.h>

// Problem constants (from reference)
#define BB 2
#define NN 8192
#define DD 64
#define KNN 11
#define TOTAL (BB * NN)           // 16384 points
#define CTILES (NN / 16)          // 512 column tiles per batch
#define GWAVES (TOTAL / 32)       // 512 waves, each owning 32 rows
#define WAVES_PER_BLOCK 4
#define LDS_STRIDE 20             // 32x16 tile rows padded to 20 floats (80B, 16B-aligned)

typedef __attribute__((ext_vector_type(2))) float v2f;
typedef __attribute__((ext_vector_type(4))) float v4f;
typedef __attribute__((ext_vector_type(8))) float v8f;

// ---------------------------------------------------------------------------
// Kernel 1: feature encoder  f = (p^T W1 + b1) W2 + b2, plus sq = ||f||^2
// ---------------------------------------------------------------------------
__global__ __launch_bounds__(64) void crf_encode(
    const float* __restrict__ p, const float* __restrict__ W1,
    const float* __restrict__ b1, const float* __restrict__ W2,
    const float* __restrict__ b2, float* __restrict__ f,
    float* __restrict__ sq)
{
    __shared__ float pv[DD];
    __shared__ float f1[DD];
    __shared__ float red[DD];

    const int t  = threadIdx.x;
    const int bn = blockIdx.x;
    const int b  = bn / NN;
    const int n  = bn - b * NN;

    pv[t] = p[(b * DD + t) * NN + n];
    __syncthreads();

    float a1 = b1[t];
    #pragma unroll 8
    for (int d = 0; d < DD; ++d) a1 = fmaf(pv[d], W1[d * DD + t], a1);
    f1[t] = a1;
    __syncthreads();

    float a2 = b2[t];
    #pragma unroll 8
    for (int d = 0; d < DD; ++d) a2 = fmaf(f1[d], W2[d * DD + t], a2);

    f[bn * DD + t] = a2;
    red[t] = a2 * a2;
    __syncthreads();
    for (int s = 32; s > 0; s >>= 1) {
        if (t < s) red[t] += red[t + s];
        __syncthreads();
    }
    if (t == 0) sq[bn] = red[0];
}

// ---------------------------------------------------------------------------
// Kernel 2: fused gram (WMMA f32 16x16x4) + per-row top-11 neighbors.
// Each wave owns 32 rows (two 16-row tiles, acc0/acc1) and sweeps the 512
// column tiles of its batch: 16 batched B-loads -> 32 interleaved WMMAs.
// d2 (32x16) goes through a private LDS slice; every lane owns one row's
// sorted 11-entry register list. Wave-local s_wait_dscnt, no block barriers.
// ---------------------------------------------------------------------------
__global__ __launch_bounds__(128) void gram_topk(
    const float* __restrict__ f, const float* __restrict__ sq,
    float* __restrict__ wOut, int* __restrict__ idxOut)
{
    const int lane  = threadIdx.x & 31;
    const int wave  = threadIdx.x >> 5;
    const int gwave = blockIdx.x * WAVES_PER_BLOCK + wave;  // [0, GWAVES)
    const int b     = gwave / (GWAVES / BB);
    const int rowBase      = gwave * 32;                    // 32 rows per wave
    const int colBatchBase = b * NN;

    __shared__ __align__(16) float lds[WAVES_PER_BLOCK][32][LDS_STRIDE];

    const int half = lane >> 4;            // f32 A/B layout halves (ISA 7.12.2)
    const int lm   = lane & 15;

    // Preload A operands for both row tiles (16 rows x 64 feats each).
    v2f a0[16], a1[16];
    const float* fr0 = f + (rowBase + lm) * DD + 2 * half;
    const float* fr1 = fr0 + 16 * DD;
    #pragma unroll
    for (int s = 0; s < 16; ++s) {
        a0[s] = *(const v2f*)(fr0 + 4 * s);
        a1[s] = *(const v2f*)(fr1 + 4 * s);
    }

    float sqRow0[8], sqRow1[8];
    #pragma unroll
    for (int r = 0; r < 8; ++r) {
        sqRow0[r] = sq[rowBase + r + 8 * half];
        sqRow1[r] = sq[rowBase + 16 + r + 8 * half];
    }

    float dist[KNN];
    int   idx[KNN];
    #pragma unroll
    for (int k = 0; k < KNN; ++k) { dist[k] = 3.4e38f; idx[k] = rowBase; }

    float* ldsRowW0 = &lds[wave][8 * half][lm];       // acc0 store base
    float* ldsRowW1 = &lds[wave][16 + 8 * half][lm];  // acc1 store base
    const float* ldsRowR = &lds[wave][lane][0];       // scan base (own row)

    for (int ct = 0; ct < CTILES; ++ct) {
        const int    colBase = colBatchBase + ct * 16;
        const float* fc      = f + (colBase + lm) * DD + 2 * half;

        // Warm next column tile into cache while this one computes.
        if (ct + 1 < CTILES)
            __builtin_prefetch(fc + 16 * DD, 0, 1);

        // Batch the 16 B-operand loads (one loadcnt wait, not 16).
        v2f bop[16];
        #pragma unroll
        for (int s = 0; s < 16; ++s)
            bop[s] = *(const v2f*)(fc + 4 * s);

        const float sqCol = sq[colBase + lm];

        // 32 WMMAs, acc0/acc1 interleaved to hide accumulate latency.
        v8f acc0 = {}, acc1 = {};
        #pragma unroll
        for (int s = 0; s < 16; ++s) {
            acc0 = __builtin_amdgcn_wmma_f32_16x16x4_f32(
                false, a0[s], false, bop[s], (short)0, acc0, false, false);
            acc1 = __builtin_amdgcn_wmma_f32_16x16x4_f32(
                false, a1[s], false, bop[s], (short)0, acc1, false, false);
        }

        // d2 = ||row||^2 + ||col||^2 - 2*S  -> LDS (C layout: VGPR r = row r / r+8)
        #pragma unroll
        for (int r = 0; r < 8; ++r) {
            ldsRowW0[r * LDS_STRIDE] = sqRow0[r] + sqCol - 2.0f * acc0[r];
            ldsRowW1[r * LDS_STRIDE] = sqRow1[r] + sqCol - 2.0f * acc1[r];
        }

        // Wave-local visibility: DS ops are in-order per wave; one dscnt wait
        // makes the 16 stores visible to this wave's cross-lane reads.
        asm volatile("s_wait_dscnt 0x0" ::: "memory");

        // Every lane scans its own row (16 values, 4x ds_load_b128).
        #pragma unroll
        for (int cc = 0; cc < 4; ++cc) {
            const v4f dv = *(const v4f*)(ldsRowR + 4 * cc);
            #pragma unroll
            for (int j = 0; j < 4; ++j) {
                const float d = dv[j];
                if (d < dist[KNN - 1]) {
                    float cd = d;
                    int   ci = colBase + 4 * cc + j;
                    #pragma unroll
                    for (int k = 0; k < KNN; ++k) {
                        if (cd < dist[k]) {
                            float td = dist[k]; dist[k] = cd; cd = td;
                            int   ti = idx[k];  idx[k]  = ci; ci = ti;
                        }
                    }
                }
            }
        }
    }

    // w = exp(-d2); idx stored flat in [0, B*N)
    const int row = rowBase + lane;
    #pragma unroll
    for (int k = 0; k < KNN; ++k) {
        wOut[row * KNN + k]   = __expf(-dist[k]);
        idxOut[row * KNN + k] = idx[k];
    }
}

// ---------------------------------------------------------------------------
// Kernel 3: one mean-field iteration  u' = logits - sum_k w_k * sigmoid(u[idx_k])
// ---------------------------------------------------------------------------
__global__ __launch_bounds__(256) void mf_iter(
    const float* __restrict__ logits, const float* __restrict__ uPrev,
    const float* __restrict__ w, const int* __restrict__ idx,
    float* __restrict__ uNext)
{
    const int i = blockIdx.x * blockDim.x + threadIdx.x;
    if (i >= TOTAL) return;
    float msg = 0.0f;
    #pragma unroll
    for (int k = 0; k < KNN; ++k) {
        const float uv = uPrev[idx[i * KNN + k]];
        msg = fmaf(w[i * KNN + k], 1.0f / (1.0f + __expf(-uv)), msg);
    }
    uNext[i] = logits[i] - msg;
}

__global__ __launch_bounds__(256) void mf_final(
    const float* __restrict__ u, float* __restrict__ out)
{
    const int i = blockIdx.x * blockDim.x + threadIdx.x;
    if (i < TOTAL) out[i] = 1.0f / (1.0f + __expf(-u[i]));
}

// ---------------------------------------------------------------------------
extern "C" void kernel_launch(void* const* d_in, const int* in_sizes, int n_in,
                              void* d_out, int out_size, void* d_ws, size_t ws_size,
                              hipStream_t stream) {
    const float* logits = (const float*)d_in[0];   // [B, N]
    const float* p      = (const float*)d_in[1];   // [B, D, N]
    const float* W1     = (const float*)d_in[2];   // [D, D]
    const float* b1     = (const float*)d_in[3];   // [D]
    const float* W2     = (const float*)d_in[4];   // [D, D]
    const float* b2     = (const float*)d_in[5];   // [D]
    float*       out    = (float*)d_out;           // [B, N]

    // Workspace carve-up (~5.9 MB)
    float* f   = (float*)d_ws;                     // TOTAL*DD
    float* sqv = f   + (size_t)TOTAL * DD;         // TOTAL
    float* wnn = sqv + TOTAL;                      // TOTAL*KNN
    float* uA  = wnn + (size_t)TOTAL * KNN;        // TOTAL
    float* uB  = uA  + TOTAL;                      // TOTAL
    int*   inn = (int*)(uB + TOTAL);               // TOTAL*KNN

    // 1) feature encoding + norms
    crf_encode<<<TOTAL, DD, 0, stream>>>(p, W1, b1, W2, b2, f, sqv);

    // 2) WMMA gram + fused top-11 neighbor selection
    gram_topk<<<GWAVES / WAVES_PER_BLOCK, WAVES_PER_BLOCK * 32, 0, stream>>>(
        f, sqv, wnn, inn);

    // 3) 5 mean-field iterations (ping-pong), u0 = logits
    const int blks = (TOTAL + 255) / 256;
    mf_iter<<<blks, 256, 0, stream>>>(logits, logits, wnn, inn, uA);
    mf_iter<<<blks, 256, 0, stream>>>(logits, uA, wnn, inn, uB);
    mf_iter<<<blks, 256, 0, stream>>>(logits, uB, wnn, inn, uA);
    mf_iter<<<blks, 256, 0, stream>>>(logits, uA, wnn, inn, uB);
    mf_iter<<<blks, 256, 0, stream>>>(logits, uB, wnn, inn, uA);

    // 4) probabilities
    mf_final<<<blks, 256, 0, stream>>>(uA, out);
}